// RegionSelection_74517682585706
// MI455X (gfx1250) — compile-verified
//
#include <hip/hip_runtime.h>

// Problem constants (fixed by the reference's setup_inputs)
#define B_   8
#define C_   256
#define HG   80
#define WG   80
#define HU   160
#define WU   160
#define HWU  (HU * WU)      // 25600 upsampled pixels per batch
#define HWG  (HG * WG)      // 6400 attention pixels per batch
#define HW4  (HWU / 4)      // 6400 float4 per (b,c) plane
#define KSEL 1920           // int(0.3 * 80 * 80)
#define ASC_RANK (HWU - 1 - KSEL)   // 23679: ascending order statistic == descending rank 1920

typedef float f32x4 __attribute__((ext_vector_type(4)));

// Low 32 bits of a flat shared-space address == LDS byte offset within the
// workgroup allocation (aperture lives in addr[63:32] on CDNA5).
__device__ __forceinline__ unsigned lds_off_of(const void* p) {
  return (unsigned)(unsigned long long)p;
}

// Bilinear upsample (scale 2, half-pixel centers, edge-clamped — exactly
// matches jax.image.resize's renormalized triangle kernel for scale 2).
__device__ __forceinline__ float up_val(const float* __restrict__ att, int s) {
  int y = s / WU;
  int x = s - y * WU;
  int ym = y >> 1, xm = x >> 1;
  int iy0, iy1, ix0, ix1;
  float wy0, wx0;
  if (y & 1) { iy0 = ym; iy1 = (ym + 1 < HG) ? ym + 1 : HG - 1; wy0 = 0.75f; }
  else       { iy0 = (ym > 0) ? ym - 1 : 0; iy1 = ym;           wy0 = 0.25f; }
  if (x & 1) { ix0 = xm; ix1 = (xm + 1 < WG) ? xm + 1 : WG - 1; wx0 = 0.75f; }
  else       { ix0 = (xm > 0) ? xm - 1 : 0; ix1 = xm;           wx0 = 0.25f; }
  float wy1 = 1.0f - wy0, wx1 = 1.0f - wx0;
  float r0 = wx0 * att[iy0 * WG + ix0] + wx1 * att[iy0 * WG + ix1];
  float r1 = wx0 * att[iy1 * WG + ix0] + wx1 * att[iy1 * WG + ix1];
  return wy0 * r0 + wy1 * r1;
}

// One workgroup per batch: stage attention tile to LDS via CDNA5 async copy
// (ASYNCcnt path), exact 4-round radix-select of the rank-1920 threshold,
// then emit mask (d_out tail) and the (mask + 0.1) broadcast factor table.
__global__ void __launch_bounds__(1024)
select_mask_kernel(const float* __restrict__ attn,
                   float* __restrict__ factor,
                   float* __restrict__ mask_out) {
  __shared__ float att[HWG];          // 25.6 KB of the 320 KB WGP LDS
  __shared__ unsigned hist[256];
  __shared__ unsigned sel_prefix;
  __shared__ unsigned sel_rank;

  const int b   = blockIdx.x;
  const int tid = threadIdx.x;
  const float* gatt = attn + (size_t)b * HWG;

  // ---- CDNA5 async global->LDS copy of the attention tile ----
  for (int i = tid; i < HWG; i += 1024) {
    unsigned loff = lds_off_of(&att[i]);
    const float* g = gatt + i;
    asm volatile("global_load_async_to_lds_b32 %0, %1, off"
                 :: "v"(loff), "v"(g) : "memory");
  }
  asm volatile("s_wait_asynccnt 0x0" ::: "memory");
  if (tid == 0) { sel_prefix = 0u; sel_rank = (unsigned)ASC_RANK; }
  __syncthreads();

  // ---- 4-round 256-bin radix select on float bit patterns (values in [0,1),
  //      so uint32 order == float order). Finds ascending rank ASC_RANK. ----
  for (int shift = 24; shift >= 0; shift -= 8) {
    if (tid < 256) hist[tid] = 0u;
    __syncthreads();
    const unsigned pfx = sel_prefix;
    const unsigned hm  = (shift == 24) ? 0u : (0xFFFFFFFFu << (shift + 8));
    for (int s = tid; s < HWU; s += 1024) {
      unsigned kk = __float_as_uint(up_val(att, s));
      if ((kk & hm) == pfx) atomicAdd(&hist[(kk >> shift) & 255u], 1u);
    }
    __syncthreads();
    if (tid == 0) {
      unsigned rr = sel_rank, cum = 0u, c;
      for (c = 0u; c < 256u; ++c) {
        unsigned h = hist[c];
        if (cum + h > rr) break;
        cum += h;
      }
      sel_prefix = pfx | (c << shift);
      sel_rank   = rr - cum;
    }
    __syncthreads();
  }

  // ---- Emit mask and the factor = mask + 0.1 table ----
  const unsigned tkey = sel_prefix;
  float* fb = factor   + (size_t)b * HWU;
  float* mb = mask_out + (size_t)b * HWU;
  for (int s = tid; s < HWU; s += 1024) {
    unsigned kk = __float_as_uint(up_val(att, s));
    float m = (kk >= tkey) ? 1.0f : 0.0f;   // == float >= for non-negative values
    mb[s] = m;
    fb[s] = m + 0.1f;
  }
}

// Fused streaming multiply: weighted[b,c,y,x] = local_feat[b,c,y,x] * factor[b,y,x].
// One block per (b,c) plane: all addressing is scalar (SALU), each thread walks
// 25 coalesced B128 chunks of its plane -> 25 independent NT loads in flight per
// wave. NT hints keep the 420 MB stream from evicting the L2-resident factor
// table (0.8 MB, re-read 256x at RT policy).
__global__ void __launch_bounds__(256)
weighted_kernel(const f32x4* __restrict__ lf,
                const f32x4* __restrict__ factor4,
                f32x4* __restrict__ out) {
  const unsigned plane = blockIdx.x;        // b*C_ + c, 0..2047 (uniform/SGPR)
  const unsigned b     = plane >> 8;        // / C_ (scalar shift)
  const f32x4* __restrict__ src  = lf      + (size_t)plane * HW4;
  const f32x4* __restrict__ fsrc = factor4 + (size_t)b     * HW4;
  f32x4* __restrict__ dst        = out     + (size_t)plane * HW4;

  const unsigned t = threadIdx.x;
#pragma unroll 5
  for (int it = 0; it < HW4 / 256; ++it) {  // 25 iterations
    unsigned s4 = (unsigned)it * 256u + t;
    f32x4 f = fsrc[s4];                               // cached (RT policy)
    f32x4 v = __builtin_nontemporal_load(&src[s4]);   // streamed, NT
    __builtin_nontemporal_store(v * f, &dst[s4]);     // streamed, NT
  }
}

extern "C" void kernel_launch(void* const* d_in, const int* in_sizes, int n_in,
                              void* d_out, int out_size, void* d_ws, size_t ws_size,
                              hipStream_t stream) {
  const float* local_feat = (const float*)d_in[0];
  const float* attn       = (const float*)d_in[1];
  // d_in[2] = spatial_scale (== 2, compile-time constant here)

  float* out      = (float*)d_out;
  float* mask_out = out + (size_t)B_ * C_ * HWU;   // second tuple element
  float* factor   = (float*)d_ws;                  // B_*HWU floats = 819,200 B

  select_mask_kernel<<<dim3(B_), dim3(1024), 0, stream>>>(attn, factor, mask_out);

  weighted_kernel<<<dim3(B_ * C_), dim3(256), 0, stream>>>(
      (const f32x4*)local_feat, (const f32x4*)factor, (f32x4*)out);
}